// GraphNN_42408507081109
// MI455X (gfx1250) — compile-verified
//
#include <hip/hip_runtime.h>
#include <hip/hip_bf16.h>
#include <math.h>

// ---------------------------------------------------------------------------
// GraphNN pipeline for MI455X (gfx1250, wave32).
//   k1: zero padded agg buffer (N x 8 f32)
//   k2: edge scatter-add  agg[dst] += x[src] * edge_attr[:,1]   (atomics)
//   k3: h = tanh(agg@Wrel^T + x@Wroot^T + b)  via V_WMMA_F32_16X16X4_F32
//   k4: per-block counts of valid first-half edges
//   k5: exclusive scan of block counts (single block)
//   k6: compact valid edges, 33-dim scoring dot, 2-way argmin, write outputs
// ---------------------------------------------------------------------------

typedef float v2f __attribute__((ext_vector_type(2)));
typedef float v8f __attribute__((ext_vector_type(8)));

#define TPB 256

// --- k1: zero scratch ------------------------------------------------------
__global__ void k_zero_f32(float* __restrict__ p, int n) {
  int i = blockIdx.x * blockDim.x + threadIdx.x;
  if (i < n) p[i] = 0.0f;
}

// --- k2: edge scatter ------------------------------------------------------
// edges flat: src = edges[i], dst = edges[E + i], i in [0, E)
// edge_attr row-major (E,2): w = eattr[2*i+1]
__global__ void k_edge_scatter(const int* __restrict__ edges,
                               const float* __restrict__ x,
                               const float* __restrict__ eattr,
                               float* __restrict__ agg,  // N x 8, cols 5..7 stay 0
                               int E) {
  int i = blockIdx.x * blockDim.x + threadIdx.x;
  if (i >= E) return;
  int src = edges[i];
  int dst = edges[E + i];
  float w = eattr[2 * i + 1];
#pragma unroll
  for (int k = 0; k < 5; ++k) {
    atomicAdd(&agg[dst * 8 + k], x[src * 5 + k] * w);
  }
}

// --- k3: node MLP via f32 WMMA --------------------------------------------
// One wave (32 lanes) computes one 16-node x 16-hid tile:
//   C += A(agg 16x4) x B(Wrel^T 4x16)   (2 steps, K padded 5->8)
//   C += A(x   16x4) x B(Wroot^T 4x16)  (2 steps)
// A layout (32-bit 16x4): lane m=lane&15 holds row M=m; v0:K=2*half(+4s), v1:K+1
// D layout: VGPR i -> M = i + 8*(lane>>4), N = lane&15
__global__ void k_node_mlp_wmma(const float* __restrict__ agg,   // N x 8
                                const float* __restrict__ x,     // N x 5
                                const float* __restrict__ Wrel,  // 16 x 5 row-major
                                const float* __restrict__ brel,  // 16
                                const float* __restrict__ Wroot, // 16 x 5
                                float* __restrict__ h,           // N x 16
                                int nTiles, int N) {
  int wave = (int)((blockIdx.x * blockDim.x + threadIdx.x) >> 5);
  int lane = (int)(threadIdx.x & 31);
  if (wave >= nTiles) return;  // uniform per-wave branch; EXEC stays all-1s

  int m     = lane & 15;
  int halfL = lane >> 4;
  int rowA  = wave * 16 + m;
  if (rowA >= N) rowA = N - 1;  // clamp loads; garbage rows never stored

  v8f c = {};

  // agg @ Wrel^T, K = 0..7 (agg cols 5..7 are zero-padded in the buffer)
#pragma unroll
  for (int s = 0; s < 2; ++s) {
    int k0 = 4 * s + 2 * halfL;
    v2f a, b;
    a.x = agg[rowA * 8 + k0];
    a.y = agg[rowA * 8 + k0 + 1];
    b.x = (k0     < 5) ? Wrel[m * 5 + k0]     : 0.0f;  // B[k][n] = Wrel[n][k]
    b.y = (k0 + 1 < 5) ? Wrel[m * 5 + k0 + 1] : 0.0f;
    c = __builtin_amdgcn_wmma_f32_16x16x4_f32(false, a, false, b,
                                              (short)0, c, false, false);
  }
  // x @ Wroot^T, K = 0..7 (x has 5 cols; pad in-register)
#pragma unroll
  for (int s = 0; s < 2; ++s) {
    int k0 = 4 * s + 2 * halfL;
    v2f a, b;
    a.x = (k0     < 5) ? x[rowA * 5 + k0]      : 0.0f;
    a.y = (k0 + 1 < 5) ? x[rowA * 5 + k0 + 1]  : 0.0f;
    b.x = (k0     < 5) ? Wroot[m * 5 + k0]     : 0.0f;
    b.y = (k0 + 1 < 5) ? Wroot[m * 5 + k0 + 1] : 0.0f;
    c = __builtin_amdgcn_wmma_f32_16x16x4_f32(false, a, false, b,
                                              (short)0, c, false, false);
  }

  int ncol  = m;
  float bias = brel[ncol];
#pragma unroll
  for (int i = 0; i < 8; ++i) {
    int M = i + 8 * halfL;
    int node = wave * 16 + M;
    if (node < N) h[node * 16 + ncol] = tanhf(c[i] + bias);
  }
}

// --- k4: per-block valid counts -------------------------------------------
__global__ void k_valid_count(const int* __restrict__ edges,
                              const unsigned char* __restrict__ labels,
                              int* __restrict__ counts, int Ehalf, int E) {
  __shared__ int sh[TPB];
  int i = blockIdx.x * TPB + threadIdx.x;
  int v = 0;
  if (i < Ehalf) {
    int s = edges[i], d = edges[E + i];
    v = (labels[s] != 0 && labels[d] != 0) ? 1 : 0;
  }
  sh[threadIdx.x] = v;
  __syncthreads();
  for (int off = TPB / 2; off > 0; off >>= 1) {
    if ((int)threadIdx.x < off) sh[threadIdx.x] += sh[threadIdx.x + off];
    __syncthreads();
  }
  if (threadIdx.x == 0) counts[blockIdx.x] = sh[0];
}

// --- k5: exclusive scan of block counts (1 block) --------------------------
__global__ void k_scan_offsets(const int* __restrict__ counts,
                               int* __restrict__ offsets, int nb) {
  __shared__ int sh[TPB];
  int carry = 0;
  for (int base = 0; base < nb; base += TPB) {
    int i = base + (int)threadIdx.x;
    int v = (i < nb) ? counts[i] : 0;
    sh[threadIdx.x] = v;
    __syncthreads();
    for (int off = 1; off < TPB; off <<= 1) {
      int t = ((int)threadIdx.x >= off) ? sh[threadIdx.x - off] : 0;
      __syncthreads();
      sh[threadIdx.x] += t;
      __syncthreads();
    }
    int incl = sh[threadIdx.x];
    if (i < nb) offsets[i] = carry + incl - v;
    int total = sh[TPB - 1];
    __syncthreads();  // protect sh before next chunk
    carry += total;
  }
}

// --- k6: compact + score + 2-way argmin ------------------------------------
// score(copy c) = b_lin + Wlin[0:16].h[src] + Wlin[17:33].h[dst] + Wlin[16]*ea_c[0]
// The two copies share the base dot; argmin reduces to one compare (tie -> copy0).
__global__ void k_select_score(const int* __restrict__ edges,
                               const unsigned char* __restrict__ labels,
                               const float* __restrict__ eattr,
                               const float* __restrict__ h,
                               const float* __restrict__ Wlin,  // 33
                               const float* __restrict__ blin,  // 1
                               const int* __restrict__ blockOffsets,
                               float* __restrict__ out, int halfOut,
                               int Ehalf, int E) {
  __shared__ int sh[TPB];
  int i = blockIdx.x * TPB + threadIdx.x;
  int valid = 0, s = 0, d = 0;
  if (i < Ehalf) {
    s = edges[i];
    d = edges[E + i];
    valid = (labels[s] != 0 && labels[d] != 0) ? 1 : 0;
  }
  // intra-block inclusive scan (Hillis-Steele) -> stable compaction position
  sh[threadIdx.x] = valid;
  __syncthreads();
  for (int off = 1; off < TPB; off <<= 1) {
    int t = ((int)threadIdx.x >= off) ? sh[threadIdx.x - off] : 0;
    __syncthreads();
    sh[threadIdx.x] += t;
    __syncthreads();
  }
  int p = blockOffsets[blockIdx.x] + sh[threadIdx.x] - valid;

  if (valid) {
    // prefetch far-half attrs while the 32-float gather dot runs
    __builtin_prefetch(&eattr[2 * (i + Ehalf)], 0, 0);
    float base = blin[0];
#pragma unroll
    for (int k = 0; k < 16; ++k) base += Wlin[k] * h[s * 16 + k];
#pragma unroll
    for (int k = 0; k < 16; ++k) base += Wlin[17 + k] * h[d * 16 + k];
    float wm   = Wlin[16];
    float s0   = base + wm * eattr[2 * i];
    float s1   = base + wm * eattr[2 * (i + Ehalf)];
    float cls0 = eattr[2 * i + 1];
    float cls1 = eattr[2 * (i + Ehalf) + 1];
    int pick1  = (s1 < s0) ? 1 : 0;  // argmin: first row wins ties
    float sc   = pick1 ? s1 : s0;
    float cl   = pick1 ? cls1 : cls0;
    if (p < halfOut) {
      out[p]               = (float)s;   // e[0]
      out[halfOut + p]     = (float)d;   // e[1]
      out[2 * halfOut + p] = sc;         // min score
      out[3 * halfOut + p] = cl;         // class of argmin copy
    }
  }
}

// ---------------------------------------------------------------------------
extern "C" void kernel_launch(void* const* d_in, const int* in_sizes, int n_in,
                              void* d_out, int out_size, void* d_ws, size_t ws_size,
                              hipStream_t stream) {
  const float* x           = (const float*)d_in[0];          // (N,5)
  const int* edges         = (const int*)d_in[1];            // (2,E)
  const float* eattr       = (const float*)d_in[2];          // (E,2)
  const unsigned char* lab = (const unsigned char*)d_in[3];  // (N,) bool (1B)
  const float* Wrel        = (const float*)d_in[4];          // (16,5)
  const float* brel        = (const float*)d_in[5];          // (16,)
  const float* Wroot       = (const float*)d_in[6];          // (16,5)
  const float* Wlin        = (const float*)d_in[7];          // (1,33)
  const float* blin        = (const float*)d_in[8];          // (1,)

  const int N     = in_sizes[0] / 5;
  const int E     = in_sizes[1] / 2;   // total edges (duplicated list)
  const int Ehalf = E / 2;
  const int halfOut = out_size / 4;    // outputs: 2*half ints + half + half

  // workspace layout (256B aligned slices)
  char* ws = (char*)d_ws;
  size_t off = 0;
  auto carve = [&](size_t bytes) {
    char* p = ws + off;
    off += (bytes + 255) & ~(size_t)255;
    return p;
  };
  float* agg   = (float*)carve((size_t)N * 8 * sizeof(float));  // padded N x 8
  float* hbuf  = (float*)carve((size_t)N * 16 * sizeof(float)); // N x 16
  const int nbScan = (Ehalf + TPB - 1) / TPB;
  int* counts  = (int*)carve((size_t)nbScan * sizeof(int));
  int* offsets = (int*)carve((size_t)nbScan * sizeof(int));
  (void)ws_size;

  // k1: zero agg
  {
    int n = N * 8;
    k_zero_f32<<<(n + TPB - 1) / TPB, TPB, 0, stream>>>(agg, n);
  }
  // k2: scatter
  k_edge_scatter<<<(E + TPB - 1) / TPB, TPB, 0, stream>>>(edges, x, eattr, agg, E);
  // k3: WMMA node MLP (8 waves/block, 1 tile of 16 nodes per wave)
  {
    int nTiles = (N + 15) / 16;
    int wavesPerBlock = TPB / 32;
    int blocks = (nTiles + wavesPerBlock - 1) / wavesPerBlock;
    k_node_mlp_wmma<<<blocks, TPB, 0, stream>>>(agg, x, Wrel, brel, Wroot,
                                                hbuf, nTiles, N);
  }
  // k4: valid counts
  k_valid_count<<<nbScan, TPB, 0, stream>>>(edges, lab, counts, Ehalf, E);
  // k5: scan (single block)
  k_scan_offsets<<<1, TPB, 0, stream>>>(counts, offsets, nbScan);
  // k6: compact + score
  k_select_score<<<nbScan, TPB, 0, stream>>>(edges, lab, eattr, hbuf, Wlin, blin,
                                             offsets, (float*)d_out, halfOut,
                                             Ehalf, E);
}